// ProbChamferLoss_77214922047597
// MI455X (gfx1250) — compile-verified
//
#include <hip/hip_runtime.h>
#include <hip/hip_bf16.h>
#include <math.h>

typedef __attribute__((ext_vector_type(2))) float v2f;
typedef __attribute__((ext_vector_type(8))) float v8f;

#define TS 16
#define ROWS_PER_BLOCK 128
#define COLS_PER_BLOCK 2048
#define NT (COLS_PER_BLOCK / TS)
#define THREADS 256

__device__ __forceinline__ unsigned long long pack_key(float d2, unsigned idx) {
  // d2 >= 0 up to rounding; clamp so float-bit ordering == value ordering.
  return ((unsigned long long)__float_as_uint(fmaxf(d2, 0.0f)) << 32) |
         (unsigned long long)idx;
}

__device__ __forceinline__ unsigned long long umin64(unsigned long long a,
                                                     unsigned long long b) {
  return a < b ? a : b;
}

// ---------------------------------------------------------------------------
// Kernel 1: build augmented matrices, reset key arrays, zero the output.
//   A4[i] = (-2rx, -2ry, -2rz, 1),  rn[i] = |r_i|^2
//   B4[j] = (kx, ky, kz, |k_j|^2)   with k = R*src + t
// WMMA then yields  acc[i][j] = -2 r_i . k_j + |k_j|^2 ;  d2 = acc + rn[i].
// ---------------------------------------------------------------------------
__global__ void prep_kernel(const float* __restrict__ ref,
                            const float* __restrict__ src,
                            const float* __restrict__ gt,
                            float4* __restrict__ A4, float4* __restrict__ B4,
                            float* __restrict__ rn,
                            unsigned long long* __restrict__ fkey,
                            unsigned long long* __restrict__ bkey,
                            float* __restrict__ out, int N, int M) {
  int i = blockIdx.x * blockDim.x + threadIdx.x;
  if (i == 0) out[0] = 0.0f;
  if (i < N) {
    float x = ref[3 * i + 0], y = ref[3 * i + 1], z = ref[3 * i + 2];
    A4[i] = make_float4(-2.0f * x, -2.0f * y, -2.0f * z, 1.0f);
    rn[i] = x * x + y * y + z * z;
    fkey[i] = ~0ull;
  }
  if (i < M) {
    float sx = src[3 * i + 0], sy = src[3 * i + 1], sz = src[3 * i + 2];
    float kx = gt[0] * sx + gt[1] * sy + gt[2]  * sz + gt[3];
    float ky = gt[4] * sx + gt[5] * sy + gt[6]  * sz + gt[7];
    float kz = gt[8] * sx + gt[9] * sy + gt[10] * sz + gt[11];
    B4[i] = make_float4(kx, ky, kz, kx * kx + ky * ky + kz * kz);
    bkey[i] = ~0ull;
  }
}

// ---------------------------------------------------------------------------
// Kernel 2: fused distance GEMM (V_WMMA_F32_16X16X4_F32) + min/argmin.
// Block = 8 waves; wave w owns rows [bx*128 + w*16, +16), cols [by*2048,+2048).
// B fragments are software-pipelined b64 loads; backward column mins merge via
// DS_MIN_U64 into a block-local key table, forward row mins via shfl-xor tree.
// ---------------------------------------------------------------------------
__global__ void __launch_bounds__(THREADS)
dist_kernel(const float2* __restrict__ A2, const float2* __restrict__ B2,
            const float* __restrict__ rn,
            unsigned long long* __restrict__ fkey,
            unsigned long long* __restrict__ bkey, int N, int M) {
  __shared__ unsigned long long bwd_lds[COLS_PER_BLOCK];

  const int lane = threadIdx.x & 31;
  const int wave = threadIdx.x >> 5;
  const int half = lane >> 4;   // 0: lanes 0-15 (K0,K1), 1: lanes 16-31 (K2,K3)
  const int l16  = lane & 15;

  for (int c = threadIdx.x; c < COLS_PER_BLOCK; c += THREADS) bwd_lds[c] = ~0ull;
  __syncthreads();

  const int row_base = blockIdx.x * ROWS_PER_BLOCK + wave * TS;
  const int col_base = blockIdx.y * COLS_PER_BLOCK;

  // A fragment (16x4 fp32): lane l16 = row; half selects which K pair (b64).
  const float2 af = A2[2 * (row_base + l16) + half];
  v2f afrag;
  afrag.x = af.x;
  afrag.y = af.y;

  // Row norms for the 8 C/D rows this lane touches (M = v + 8*half).
  float rnv[8];
#pragma unroll
  for (int v = 0; v < 8; ++v) rnv[v] = rn[row_base + v + 8 * half];

  // Running forward (row-wise) packed min keys.
  unsigned long long fk[8];
#pragma unroll
  for (int v = 0; v < 8; ++v) fk[v] = ~0ull;

  // Software-pipelined B fragment loads: fetch tile ct+1 while crunching ct.
  const int colb = col_base + l16;
  float2 bv = B2[2 * colb + half];

  for (int ct = 0; ct < NT; ++ct) {
    const int ctn = (ct + 1 < NT) ? (ct + 1) : 0;   // scalar clamp, no divergence
    const float2 bnext = B2[2 * (colb + ctn * TS) + half];

    v2f bfrag;
    bfrag.x = bv.x;
    bfrag.y = bv.y;

    v8f acc = {};
    acc = __builtin_amdgcn_wmma_f32_16x16x4_f32(
        /*neg_a=*/false, afrag, /*neg_b=*/false, bfrag,
        /*c_mod=*/(short)0, acc, /*reuse_a=*/false, /*reuse_b=*/false);

    const int col = colb + ct * TS;
    unsigned long long bk = ~0ull;  // column-wise min over this lane's 8 rows
#pragma unroll
    for (int v = 0; v < 8; ++v) {
      const float d2 = acc[v] + rnv[v];
      fk[v] = umin64(fk[v], pack_key(d2, (unsigned)col));
      bk    = umin64(bk, pack_key(d2, (unsigned)(row_base + v + 8 * half)));
    }
    // Lanes l and l+16 share a column: both issue DS_MIN_U64 (2-way serialize
    // in the DS atomic unit beats swizzle + exec-mask branching).
    atomicMin(&bwd_lds[ct * TS + l16], bk);

    bv = bnext;
  }

  // Forward reduction: 16 lanes of each half share a row per VGPR index.
#pragma unroll
  for (int v = 0; v < 8; ++v) {
    unsigned long long k = fk[v];
#pragma unroll
    for (int m = 8; m >= 1; m >>= 1) {
      const unsigned long long o =
          (unsigned long long)__shfl_xor((long long)k, m, 32);
      k = umin64(k, o);
    }
    if (l16 == 0) atomicMin(&fkey[row_base + v + 8 * half], k);
  }

  __syncthreads();
  for (int c = threadIdx.x; c < COLS_PER_BLOCK; c += THREADS)
    atomicMin(&bkey[col_base + c], bwd_lds[c]);
}

// ---------------------------------------------------------------------------
// Kernel 3: unpack keys -> per-point loss terms -> scalar.
// ---------------------------------------------------------------------------
__global__ void loss_kernel(const unsigned long long* __restrict__ fkey,
                            const unsigned long long* __restrict__ bkey,
                            const float* __restrict__ ref_sigma,
                            const float* __restrict__ src_sigma,
                            float* __restrict__ out, int N, int M) {
  const int t = blockIdx.x * blockDim.x + threadIdx.x;
  float term = 0.0f;
  if (t < N) {
    const unsigned long long k = fkey[t];
    const float d = sqrtf(__uint_as_float((unsigned)(k >> 32)));
    const unsigned j = (unsigned)(k & 0xffffffffu);
    const float s = 0.5f * (ref_sigma[t] + src_sigma[j]);
    term = (logf(s) + d / s) * (1.0f / (float)N);
  } else if (t < N + M) {
    const int c = t - N;
    const unsigned long long k = bkey[c];
    const float d = sqrtf(__uint_as_float((unsigned)(k >> 32)));
    const unsigned i = (unsigned)(k & 0xffffffffu);
    const float s = 0.5f * (src_sigma[c] + ref_sigma[i]);
    term = (logf(s) + d / s) * (1.0f / (float)M);
  }
#pragma unroll
  for (int off = 16; off >= 1; off >>= 1) term += __shfl_down(term, off, 32);
  if ((threadIdx.x & 31) == 0) atomicAdd(out, term);
}

// ---------------------------------------------------------------------------
extern "C" void kernel_launch(void* const* d_in, const int* in_sizes, int n_in,
                              void* d_out, int out_size, void* d_ws,
                              size_t ws_size, hipStream_t stream) {
  const float* ref = (const float*)d_in[0];   // (N,3) f32
  const float* src = (const float*)d_in[1];   // (M,3) f32
  const float* gt  = (const float*)d_in[2];   // (4,4) f32 row-major
  const float* rs  = (const float*)d_in[3];   // (N,)  f32
  const float* ss  = (const float*)d_in[4];   // (M,)  f32
  float* out = (float*)d_out;

  const int N = in_sizes[0] / 3;
  const int M = in_sizes[1] / 3;

  // Workspace carve-out: A4|B4|rn|fkey|bkey  (~416 KB for 8192/8192)
  float4* A4 = (float4*)d_ws;
  float4* B4 = A4 + N;
  float*  rn = (float*)(B4 + M);
  unsigned long long* fk = (unsigned long long*)(rn + N);
  unsigned long long* bk = fk + N;

  const int P = (N > M ? N : M);
  prep_kernel<<<(P + 255) / 256, 256, 0, stream>>>(ref, src, gt, A4, B4, rn,
                                                   fk, bk, out, N, M);

  dim3 grid(N / ROWS_PER_BLOCK, M / COLS_PER_BLOCK);  // 64 x 4 for 8192^2
  dist_kernel<<<grid, THREADS, 0, stream>>>((const float2*)A4,
                                            (const float2*)B4, rn, fk, bk, N,
                                            M);

  loss_kernel<<<(N + M + 255) / 256, 256, 0, stream>>>(fk, bk, rs, ss, out, N,
                                                       M);
}